// MultiViewDGT_32272384262973
// MI455X (gfx1250) — compile-verified
//
#include <hip/hip_runtime.h>
#include <hip/hip_bf16.h>

#define DEVINL __device__ __forceinline__

static constexpr int N_   = 20000;
static constexpr int XD_  = 64;
static constexpr int HID_ = 128;
static constexpr int NH_  = 4;
static constexpr int V_   = 4;
static constexpr int E_   = 160000;
static constexpr int B_   = 1024;
static constexpr int MKT_ = 32;
static constexpr int P_   = 512;
static constexpr int L_   = 64;
static constexpr int C_   = 128;
static constexpr int HC_  = 512;
#define INV_SQRT_C 0.08838834764831845f   // 1/sqrt(128)

typedef float v2f __attribute__((ext_vector_type(2)));
typedef float v8f __attribute__((ext_vector_type(8)));

DEVINL unsigned f2ord(float f) {
    unsigned u = __float_as_uint(f);
    return (u & 0x80000000u) ? ~u : (u | 0x80000000u);
}
DEVINL float ord2f(unsigned k) {
    return (k & 0x80000000u) ? __uint_as_float(k & 0x7fffffffu) : __uint_as_float(~k);
}
DEVINL float sigm(float x) { return 1.0f / (1.0f + __expf(-x)); }

// ---------------------------------------------------------------------------
// WMMA fp32 GEMM, register-blocked + software-pipelined:
// one wave32 computes a 16x64 output tile (4 x v8f accumulators) via
// V_WMMA_F32_16X16X4_F32.  A fragment reused across 4 N-subtiles; fragments
// for step k+1 are loaded (double-buffered) while step k's WMMAs execute.
// Requires Nn % 64 == 0 and K % 4 == 0 (true for all GEMMs here).
// A layout: lanes 0-15 -> {A[r][k0],A[r][k0+1]}, lanes 16-31 -> {A[r][k0+2],A[r][k0+3]}
// B layout mirrored (lane = column).  C/D: vgpr r, lanes0-15 M=r, lanes16-31 M=r+8.
// ---------------------------------------------------------------------------
__global__ void gemm_wmma_f32(const float* __restrict__ A, const float* __restrict__ Bm,
                              const float* __restrict__ bias, float* __restrict__ D,
                              int M, int K, int Nn, int relu)
{
    int wave = (blockIdx.x * blockDim.x + threadIdx.x) >> 5;
    int lane = threadIdx.x & 31;
    int nt64 = Nn >> 6;
    int tm = wave / nt64;
    int tn = wave - tm * nt64;
    if (tm * 16 >= M) return;

    int row   = tm * 16 + (lane & 15);
    int colb  = tn * 64 + (lane & 15);
    int khalf = (lane >> 4) * 2;   // 0 or 2

    const float* ap = A + (size_t)row * K + khalf;
    const float* bp = Bm + (size_t)khalf * Nn + colb;

    __builtin_prefetch(ap, 0, 1);
    __builtin_prefetch(bp, 0, 1);

    // prologue loads (step 0)
    v2f a, b[4];
    a.x = ap[0]; a.y = ap[1];
#pragma unroll
    for (int t = 0; t < 4; ++t) { b[t].x = bp[t * 16]; b[t].y = bp[Nn + t * 16]; }

    v8f acc[4] = {{}, {}, {}, {}};
    for (int k0 = 0; k0 < K - 4; k0 += 4) {
        ap += 4;
        bp += (size_t)4 * Nn;
        // issue next-step loads before consuming current fragments
        v2f an, bn[4];
        an.x = ap[0]; an.y = ap[1];
#pragma unroll
        for (int t = 0; t < 4; ++t) { bn[t].x = bp[t * 16]; bn[t].y = bp[Nn + t * 16]; }
#pragma unroll
        for (int t = 0; t < 4; ++t)
            acc[t] = __builtin_amdgcn_wmma_f32_16x16x4_f32(false, a, false, b[t],
                                                           (short)0, acc[t], false, false);
        a = an;
#pragma unroll
        for (int t = 0; t < 4; ++t) b[t] = bn[t];
    }
    // epilogue (last K-step)
#pragma unroll
    for (int t = 0; t < 4; ++t)
        acc[t] = __builtin_amdgcn_wmma_f32_16x16x4_f32(false, a, false, b[t],
                                                       (short)0, acc[t], false, false);

    int rbase = (lane >> 4) * 8;
#pragma unroll
    for (int t = 0; t < 4; ++t) {
        int col = colb + t * 16;
        float bv = bias[col];
#pragma unroll
        for (int r = 0; r < 8; ++r) {
            int orow = tm * 16 + rbase + r;
            float val = acc[t][r] + bv;
            if (relu) val = fmaxf(val, 0.0f);
            D[(size_t)orow * Nn + col] = val;
        }
    }
}

// ---------------------------------------------------------------------------
// Edge-weight per-view normalization (views 1..3: (x-mean)/(std+1e-6), biased std)
// ---------------------------------------------------------------------------
__global__ void ew_stats(const float* __restrict__ ew, float* __restrict__ stats)
{
    int vi = blockIdx.x;
    __shared__ float ss[256], sq[256];
    float s = 0.f, q2 = 0.f;
    for (int i = threadIdx.x; i < E_; i += blockDim.x) {
        float x = ew[(size_t)vi * E_ + i];
        s += x; q2 += x * x;
    }
    ss[threadIdx.x] = s; sq[threadIdx.x] = q2;
    __syncthreads();
    for (int st = 128; st > 0; st >>= 1) {
        if (threadIdx.x < st) { ss[threadIdx.x] += ss[threadIdx.x + st]; sq[threadIdx.x] += sq[threadIdx.x + st]; }
        __syncthreads();
    }
    if (threadIdx.x == 0) {
        float m = ss[0] / (float)E_;
        float var = fmaxf(sq[0] / (float)E_ - m * m, 0.f);
        stats[vi * 2 + 0] = m;
        stats[vi * 2 + 1] = 1.0f / (sqrtf(var) + 1e-6f);
    }
}

__global__ void ew_apply(const float* __restrict__ ew, const float* __restrict__ stats,
                         float* __restrict__ ewn, int total)
{
    int i = blockIdx.x * blockDim.x + threadIdx.x;
    if (i >= total) return;
    int vi = i / E_;
    float x = ew[i];
    ewn[i] = (vi == 0) ? x : (x - stats[vi * 2]) * stats[vi * 2 + 1];
}

// ---------------------------------------------------------------------------
// Per-tconv init: agg=0, amax=ordered(-inf)=0, den=0
// ---------------------------------------------------------------------------
__global__ void init_tconv(float* __restrict__ agg, unsigned* __restrict__ amax,
                           float* __restrict__ den, int nAgg, int nNH)
{
    int i = blockIdx.x * blockDim.x + threadIdx.x;
    if (i < nAgg) agg[i] = 0.0f;
    if (i < nNH) { amax[i] = 0u; den[i] = 0.0f; }
}

// ---------------------------------------------------------------------------
// Pass 1: a[e,h] = dot_C(q[dst], k[src] + ew*We) / sqrt(C); atomic segment-max
// One wave per edge; lanes [8h .. 8h+7] handle head h, 16 channels each.
// We (512 floats) staged in LDS per block.
// ---------------------------------------------------------------------------
__global__ void edge_logits(const float* __restrict__ q, const float* __restrict__ k,
                            const float* __restrict__ We, const float* __restrict__ ewn,
                            const int* __restrict__ src, const int* __restrict__ dst,
                            float* __restrict__ abuf, unsigned* __restrict__ amax, int nE)
{
    __shared__ float sWe[HC_];
    for (int i = threadIdx.x; i < HC_; i += blockDim.x) sWe[i] = We[i];
    __syncthreads();

    int gw = (blockIdx.x * blockDim.x + threadIdx.x) >> 5;
    if (gw >= nE) return;
    int lane = threadIdx.x & 31;
    int s = src[gw], d = dst[gw];
    float w = ewn[gw];
    int head  = lane >> 3;
    int cbase = (lane & 7) * 16;
    const float* qp = q + (size_t)d * HC_ + head * C_ + cbase;
    const float* kp = k + (size_t)s * HC_ + head * C_ + cbase;
    const float* wp = sWe + head * C_ + cbase;
    float sum = 0.f;
#pragma unroll
    for (int j = 0; j < 16; ++j) sum += qp[j] * (kp[j] + w * wp[j]);
    sum += __shfl_xor(sum, 1, 32);
    sum += __shfl_xor(sum, 2, 32);
    sum += __shfl_xor(sum, 4, 32);
    float aval = sum * INV_SQRT_C;
    if ((lane & 7) == 0) {
        abuf[(size_t)gw * NH_ + head] = aval;
        atomicMax(&amax[d * NH_ + head], f2ord(aval));
    }
}

// Pass 2: a = exp(a - amax[dst]); den[dst] += a
__global__ void edge_exp(const int* __restrict__ dst, float* __restrict__ abuf,
                         const unsigned* __restrict__ amax, float* __restrict__ den, int total)
{
    int i = blockIdx.x * blockDim.x + threadIdx.x;
    if (i >= total) return;
    int e = i >> 2, h = i & 3;
    int d = dst[e];
    float am = ord2f(amax[d * NH_ + h]);
    float ev = __expf(abuf[i] - am);
    abuf[i] = ev;
    atomicAdd(&den[d * NH_ + h], ev);
}

// Pass 3: agg[dst,c] += (1/NH) * sum_h (a/(den+eps)) * (v[src,h,c] + ew*We[h,c])
__global__ void edge_accum(const float* __restrict__ vmat, const float* __restrict__ We,
                           const float* __restrict__ ewn,
                           const int* __restrict__ src, const int* __restrict__ dst,
                           const float* __restrict__ abuf, const float* __restrict__ den,
                           float* __restrict__ agg, int nE)
{
    __shared__ float sWe[HC_];
    for (int i = threadIdx.x; i < HC_; i += blockDim.x) sWe[i] = We[i];
    __syncthreads();

    int gw = (blockIdx.x * blockDim.x + threadIdx.x) >> 5;
    if (gw >= nE) return;
    int lane = threadIdx.x & 31;
    int s = src[gw], d = dst[gw];
    float w = ewn[gw];
    float wg[NH_];
#pragma unroll
    for (int h = 0; h < NH_; ++h)
        wg[h] = abuf[(size_t)gw * NH_ + h] / (den[d * NH_ + h] + 1e-16f) * 0.25f;
#pragma unroll
    for (int j = 0; j < 4; ++j) {
        int c = lane + 32 * j;
        float contrib = 0.f;
#pragma unroll
        for (int h = 0; h < NH_; ++h)
            contrib += wg[h] * (vmat[(size_t)s * HC_ + h * C_ + c] + w * sWe[h * C_ + c]);
        atomicAdd(&agg[(size_t)d * HID_ + c], contrib);
    }
}

// beta = sigmoid([out, xr, out-xr] @ Wbeta); out = beta*xr + (1-beta)*out
__global__ void beta_combine(const float* __restrict__ agg, const float* __restrict__ xr,
                             const float* __restrict__ Wb, float* __restrict__ outv, int n)
{
    int gw = (blockIdx.x * blockDim.x + threadIdx.x) >> 5;
    if (gw >= n) return;
    int lane = threadIdx.x & 31;
    float oc[4], xc[4];
    float s = 0.f;
#pragma unroll
    for (int j = 0; j < 4; ++j) {
        int c = lane + 32 * j;
        oc[j] = agg[(size_t)gw * HID_ + c];
        xc[j] = xr[(size_t)gw * HID_ + c];
        s += oc[j] * (Wb[c] + Wb[2 * HID_ + c]) + xc[j] * (Wb[HID_ + c] - Wb[2 * HID_ + c]);
    }
#pragma unroll
    for (int m = 1; m < 32; m <<= 1) s += __shfl_xor(s, m, 32);
    float beta = sigm(s);
#pragma unroll
    for (int j = 0; j < 4; ++j) {
        int c = lane + 32 * j;
        outv[(size_t)gw * HID_ + c] = beta * xc[j] + (1.0f - beta) * oc[j];
    }
}

__global__ void acc_first(const float* __restrict__ h, const float* __restrict__ hs,
                          const float* __restrict__ g_logit, int gidx,
                          float* __restrict__ acc, int total)
{
    int i = blockIdx.x * blockDim.x + threadIdx.x;
    if (i >= total) return;
    float g = sigm(g_logit[gidx]);
    acc[i] = h[i] + g * hs[i];
}

__global__ void acc_add(const float* __restrict__ hv, const float* __restrict__ hs,
                        const float* __restrict__ g_logit, int gidx,
                        float* __restrict__ acc, int total)
{
    int i = blockIdx.x * blockDim.x + threadIdx.x;
    if (i >= total) return;
    float g = sigm(g_logit[gidx]);
    acc[i] += g * (hv[i] - hs[i]);
}

// LayerNorm over last dim (128), one wave per row
__global__ void layernorm_k(const float* __restrict__ in, const float* __restrict__ g,
                            const float* __restrict__ b, float* __restrict__ out, int n)
{
    int gw = (blockIdx.x * blockDim.x + threadIdx.x) >> 5;
    if (gw >= n) return;
    int lane = threadIdx.x & 31;
    float xv[4];
    float s = 0.f;
#pragma unroll
    for (int j = 0; j < 4; ++j) { xv[j] = in[(size_t)gw * HID_ + lane + 32 * j]; s += xv[j]; }
#pragma unroll
    for (int m = 1; m < 32; m <<= 1) s += __shfl_xor(s, m, 32);
    float mean = s * (1.0f / HID_);
    float vs = 0.f;
#pragma unroll
    for (int j = 0; j < 4; ++j) { float d = xv[j] - mean; vs += d * d; }
#pragma unroll
    for (int m = 1; m < 32; m <<= 1) vs += __shfl_xor(vs, m, 32);
    float inv = rsqrtf(vs * (1.0f / HID_) + 1e-5f);
#pragma unroll
    for (int j = 0; j < 4; ++j) {
        int c = lane + 32 * j;
        out[(size_t)gw * HID_ + c] = (xv[j] - mean) * inv * g[c] + b[c];
    }
}

// z[b] = concat(h2[anchor[b]], sum_l pw*h2[pnodes], zm[b])   -> [B, 384]
__global__ void build_z(const float* __restrict__ h2, const int* __restrict__ anchor,
                        const int* __restrict__ pf_gid, const int* __restrict__ pnodes,
                        const float* __restrict__ pw, const float* __restrict__ zm,
                        float* __restrict__ z, int nB)
{
    int gw = (blockIdx.x * blockDim.x + threadIdx.x) >> 5;
    if (gw >= nB) return;
    int lane = threadIdx.x & 31;
    int a = anchor[gw];
#pragma unroll
    for (int j = 0; j < 4; ++j) {
        int c = lane + 32 * j;
        z[(size_t)gw * 384 + c] = h2[(size_t)a * HID_ + c];
    }
    int gidr = pf_gid[gw];
    int gid  = min(max(gidr, 0), P_ - 1);
    float accv[4] = {0.f, 0.f, 0.f, 0.f};
    if (gidr >= 0) {
        for (int l = 0; l < L_; ++l) {
            int node = pnodes[gid * L_ + l];
            float wv = pw[gid * L_ + l];
#pragma unroll
            for (int j = 0; j < 4; ++j)
                accv[j] += wv * h2[(size_t)node * HID_ + lane + 32 * j];
        }
    }
#pragma unroll
    for (int j = 0; j < 4; ++j) {
        int c = lane + 32 * j;
        z[(size_t)gw * 384 + HID_ + c]     = accv[j];
        z[(size_t)gw * 384 + 2 * HID_ + c] = zm[(size_t)gw * HID_ + c];
    }
}

// out[b] = dot(hh[b], w2) + b2
__global__ void head_final(const float* __restrict__ hh, const float* __restrict__ w2,
                           const float* __restrict__ b2, float* __restrict__ out, int nB)
{
    int gw = (blockIdx.x * blockDim.x + threadIdx.x) >> 5;
    if (gw >= nB) return;
    int lane = threadIdx.x & 31;
    float s = 0.f;
#pragma unroll
    for (int j = 0; j < 4; ++j) {
        int c = lane + 32 * j;
        s += hh[(size_t)gw * HID_ + c] * w2[c];
    }
#pragma unroll
    for (int m = 1; m < 32; m <<= 1) s += __shfl_xor(s, m, 32);
    if (lane == 0) out[gw] = s + b2[0];
}

// ---------------------------------------------------------------------------
extern "C" void kernel_launch(void* const* d_in, const int* in_sizes, int n_in,
                              void* d_out, int out_size, void* d_ws, size_t ws_size,
                              hipStream_t stream)
{
    (void)in_sizes; (void)n_in; (void)out_size; (void)ws_size;

    const float* x      = (const float*)d_in[0];
    const int*   anchor = (const int*)  d_in[1];
    const float* mkt    = (const float*)d_in[2];
    const int*   pf_gid = (const int*)  d_in[3];
    const int*   pnodes = (const int*)  d_in[4];
    const float* pw     = (const float*)d_in[5];
    const int*   esrc   = (const int*)  d_in[6];
    const int*   edst   = (const int*)  d_in[7];
    const float* eww    = (const float*)d_in[8];
    // params (dict order)
    const float* enc_w1 = (const float*)d_in[9];
    const float* enc_b1 = (const float*)d_in[10];
    const float* enc_w2 = (const float*)d_in[11];
    const float* enc_b2 = (const float*)d_in[12];
    const float* ln0_g  = (const float*)d_in[13];
    const float* ln0_b  = (const float*)d_in[14];
    const float* ln1_g  = (const float*)d_in[15];
    const float* ln1_b  = (const float*)d_in[16];
    const float* ln2_g  = (const float*)d_in[17];
    const float* ln2_b  = (const float*)d_in[18];
    const float* Wq     = (const float*)d_in[19];
    const float* bq     = (const float*)d_in[20];
    const float* Wk     = (const float*)d_in[21];
    const float* bk     = (const float*)d_in[22];
    const float* Wv     = (const float*)d_in[23];
    const float* bv     = (const float*)d_in[24];
    const float* We     = (const float*)d_in[25];
    const float* Ws     = (const float*)d_in[26];
    const float* bs     = (const float*)d_in[27];
    const float* Wbeta  = (const float*)d_in[28];
    const float* g_logit= (const float*)d_in[29];
    const float* mkt_w1 = (const float*)d_in[30];
    const float* mkt_b1 = (const float*)d_in[31];
    const float* mkt_w2 = (const float*)d_in[32];
    const float* mkt_b2 = (const float*)d_in[33];
    const float* head_w1= (const float*)d_in[34];
    const float* head_b1= (const float*)d_in[35];
    const float* head_w2= (const float*)d_in[36];
    const float* head_b2= (const float*)d_in[37];

    float* W = (float*)d_ws;
    size_t o = 0;
    auto alloc = [&](size_t nf) { size_t r = o; o += (nf + 63) & ~(size_t)63; return r; };
    size_t oq   = alloc((size_t)N_ * HC_);
    size_t ok   = alloc((size_t)N_ * HC_);
    size_t ov   = alloc((size_t)N_ * HC_);
    size_t ohA  = alloc((size_t)N_ * HID_);
    size_t ohB  = alloc((size_t)N_ * HID_);
    size_t oacc = alloc((size_t)N_ * HID_);
    size_t ohs  = alloc((size_t)N_ * HID_);
    size_t ohv  = alloc((size_t)N_ * HID_);
    size_t oxr  = alloc((size_t)N_ * HID_);
    size_t oagg = alloc((size_t)N_ * HID_);
    size_t oab  = alloc((size_t)E_ * NH_);
    size_t oamax= alloc((size_t)N_ * NH_);
    size_t oden = alloc((size_t)N_ * NH_);
    size_t oewn = alloc((size_t)V_ * E_);
    size_t ost  = alloc((size_t)V_ * 2);
    size_t oz   = alloc((size_t)B_ * 3 * HID_);
    size_t ozm1 = alloc((size_t)B_ * HID_);
    size_t ozm2 = alloc((size_t)B_ * HID_);
    size_t ohh  = alloc((size_t)B_ * HID_);

    auto gemm = [&](const float* A, const float* Bm, const float* bias, float* D,
                    int M, int K, int Nn, int relu) {
        long waves = (long)((M + 15) / 16) * (Nn / 64);
        int blocks = (int)((waves * 32 + 255) / 256);
        gemm_wmma_f32<<<dim3(blocks), dim3(256), 0, stream>>>(A, Bm, bias, D, M, K, Nn, relu);
    };

    // --- edge weight normalization (same for both layers) ---
    ew_stats<<<dim3(V_), dim3(256), 0, stream>>>(eww, W + ost);
    ew_apply<<<dim3((V_ * E_ + 255) / 256), dim3(256), 0, stream>>>(eww, W + ost, W + oewn, V_ * E_);

    // --- encoder: LN(relu(x@w1+b1)@w2+b2) -> hA ---
    gemm(x, enc_w1, enc_b1, W + oq, N_, XD_, HID_, 1);      // reuse q buffer as tmp
    gemm(W + oq, enc_w2, enc_b2, W + oacc, N_, HID_, HID_, 0);
    layernorm_k<<<dim3((N_ * 32 + 255) / 256), dim3(256), 0, stream>>>(W + oacc, ln0_g, ln0_b, W + ohA, N_);

    // --- two graph layers ---
    for (int li = 0; li < 2; ++li) {
        const float* hin  = (li == 0) ? (W + ohA) : (W + ohB);
        float*       hout = (li == 0) ? (W + ohB) : (W + ohA);
        const float* lng  = (li == 0) ? ln1_g : ln2_g;
        const float* lnb  = (li == 0) ? ln1_b : ln2_b;
        for (int vi = 0; vi < V_; ++vi) {
            size_t pi = (size_t)(li * V_ + vi);
            const float* Wq_ = Wq + pi * HID_ * HC_;  const float* bq_ = bq + pi * HC_;
            const float* Wk_ = Wk + pi * HID_ * HC_;  const float* bk_ = bk + pi * HC_;
            const float* Wv_ = Wv + pi * HID_ * HC_;  const float* bv_ = bv + pi * HC_;
            const float* We_ = We + pi * HC_;
            const float* Ws_ = Ws + pi * HID_ * HID_; const float* bs_ = bs + pi * HID_;
            const float* Wb_ = Wbeta + pi * 3 * HID_;
            const int*   src = esrc + (size_t)vi * E_;
            const int*   dst = edst + (size_t)vi * E_;
            const float* ewn = W + oewn + (size_t)vi * E_;

            init_tconv<<<dim3((N_ * HID_ + 255) / 256), dim3(256), 0, stream>>>(
                W + oagg, (unsigned*)(W + oamax), W + oden, N_ * HID_, N_ * NH_);

            gemm(hin, Wq_, bq_, W + oq, N_, HID_, HC_, 0);
            gemm(hin, Wk_, bk_, W + ok, N_, HID_, HC_, 0);
            gemm(hin, Wv_, bv_, W + ov, N_, HID_, HC_, 0);
            gemm(hin, Ws_, bs_, W + oxr, N_, HID_, HID_, 0);

            edge_logits<<<dim3((E_ * 32 + 255) / 256), dim3(256), 0, stream>>>(
                W + oq, W + ok, We_, ewn, src, dst, W + oab, (unsigned*)(W + oamax), E_);
            edge_exp<<<dim3((E_ * NH_ + 255) / 256), dim3(256), 0, stream>>>(
                dst, W + oab, (const unsigned*)(W + oamax), W + oden, E_ * NH_);
            edge_accum<<<dim3((E_ * 32 + 255) / 256), dim3(256), 0, stream>>>(
                W + ov, We_, ewn, src, dst, W + oab, W + oden, W + oagg, E_);

            float* tout = (vi == 0) ? (W + ohs) : (W + ohv);
            beta_combine<<<dim3((N_ * 32 + 255) / 256), dim3(256), 0, stream>>>(
                W + oagg, W + oxr, Wb_, tout, N_);

            if (vi == 0)
                acc_first<<<dim3((N_ * HID_ + 255) / 256), dim3(256), 0, stream>>>(
                    hin, W + ohs, g_logit, li * V_, W + oacc, N_ * HID_);
            else
                acc_add<<<dim3((N_ * HID_ + 255) / 256), dim3(256), 0, stream>>>(
                    W + ohv, W + ohs, g_logit, li * V_ + vi, W + oacc, N_ * HID_);
        }
        layernorm_k<<<dim3((N_ * 32 + 255) / 256), dim3(256), 0, stream>>>(W + oacc, lng, lnb, hout, N_);
    }
    const float* h2 = W + ohA;   // layer 1 wrote hA

    // --- head ---
    gemm(mkt, mkt_w1, mkt_b1, W + ozm1, B_, MKT_, HID_, 1);
    gemm(W + ozm1, mkt_w2, mkt_b2, W + ozm2, B_, HID_, HID_, 0);
    build_z<<<dim3((B_ * 32 + 255) / 256), dim3(256), 0, stream>>>(
        h2, anchor, pf_gid, pnodes, pw, W + ozm2, W + oz, B_);
    gemm(W + oz, head_w1, head_b1, W + ohh, B_, 3 * HID_, HID_, 1);
    head_final<<<dim3((B_ * 32 + 255) / 256), dim3(256), 0, stream>>>(
        W + ohh, head_w2, head_b2, (float*)d_out, B_);
}